// ChamferLoss_34170759807614
// MI455X (gfx1250) — compile-verified
//
#include <hip/hip_runtime.h>
#include <math.h>

// ---------------------------------------------------------------------------
// Chamfer loss on MI455X (gfx1250), wave32 + V_WMMA_F32_16X16X4_F32.
//
// d2[m,n] = p2[m] + g2[n] - 2*p.g computed inside the WMMA:
//   A = [-2px,-2py,-2pz, p2]  (16x4, K0/K1 in lanes 0-15, K2/K3 in 16-31)
//   B = [gx,gy,gz, 1]         (4x16, same half-split)
//   C = broadcast(g2[n])      (free: one register splatted into 8 acc VGPRs)
//
// Each wave owns TWO 16-row A tiles (32 rows) so every g-load / g2 / B-select
// feeds two WMMAs -> per-tile VALU overhead is amortized 2x. Forward mins stay
// in accumulator components (xor-shuffle reduce at the end); backward mins are
// clamped >=0 (float bits monotone as u32), combined across halves with one
// ds_bpermute, folded into a 32KB LDS array with ds_min_u32, and flushed once
// per block with global_atomic_min_u32.
// ---------------------------------------------------------------------------

typedef __attribute__((ext_vector_type(2))) float v2f;
typedef __attribute__((ext_vector_type(8))) float v8f;

#define CH_EPS       1e-8f
#define CH_B         8
#define CH_D         3
#define POS_INF_BITS 0x7F800000u
#define BIG_F        3.0e38f
#define LDS_N        8192

// ---- kernel 0: deterministic re-init of workspace + output -----------------
__global__ void chamfer_init_kernel(unsigned int* __restrict__ bwd_min, int total,
                                    float* __restrict__ out) {
    int i = blockIdx.x * blockDim.x + threadIdx.x;
    if (i < total) bwd_min[i] = POS_INF_BITS;
    if (i == 0)    out[0] = 0.0f;
}

// ---- kernel 1: WMMA pairwise distances + fwd/bwd min reductions ------------
__global__ __launch_bounds__(256)
void chamfer_wmma_kernel(const float* __restrict__ P,   // (B,3,M) component-major
                         const float* __restrict__ G,   // (B,3,N) component-major
                         int M, int N,
                         unsigned int* __restrict__ bwd_min,  // (B,N) u32 float bits
                         float* __restrict__ out, float inv_bm) {
    __shared__ unsigned int lmin[LDS_N];          // 32 KB block-level bwd mins

    const int strips = M >> 8;                    // 256 rows per block (8 waves x 32)
    const int b      = blockIdx.x / strips;
    const int strip  = blockIdx.x % strips;
    const int tid    = threadIdx.x;
    const int lane   = tid & 31;
    const int wave   = tid >> 5;
    const int half   = lane >> 4;                 // 0: lanes 0-15, 1: lanes 16-31
    const int l16    = lane & 15;

    for (int i = tid; i < N; i += 256) lmin[i] = POS_INF_BITS;
    __syncthreads();

    // ---- two A tiles: rows m0..m0+15 and m0+16..m0+31 ----------------------
    const int    m0 = strip * 256 + wave * 32;
    const float* Pb = P + (size_t)b * CH_D * M;

    v2f a0, a1;
    {
        const int   m  = m0 + l16;
        const float px = Pb[m], py = Pb[M + m], pz = Pb[2 * M + m];
        const float p2 = px * px + py * py + pz * pz;
        a0.x = half ? (-2.0f * pz) : (-2.0f * px);
        a0.y = half ? p2           : (-2.0f * py);
    }
    {
        const int   m  = m0 + 16 + l16;
        const float px = Pb[m], py = Pb[M + m], pz = Pb[2 * M + m];
        const float p2 = px * px + py * py + pz * pz;
        a1.x = half ? (-2.0f * pz) : (-2.0f * px);
        a1.y = half ? p2           : (-2.0f * py);
    }

    v8f fmin0, fmin1;
#pragma unroll
    for (int r = 0; r < 8; ++r) { fmin0[r] = BIG_F; fmin1[r] = BIG_F; }

    const float* Gb = G + (size_t)b * CH_D * N;

#pragma unroll 2
    for (int n0 = 0; n0 < N; n0 += 16) {
        const int   n  = n0 + l16;
        const float gx = Gb[n];
        const float gy = Gb[N + n];
        const float gz = Gb[2 * N + n];
        const float g2 = gx * gx + gy * gy + gz * gz;

        v2f bm;                                   // shared B operand for both tiles
        bm.x = half ? gz   : gx;
        bm.y = half ? 1.0f : gy;

        v8f c0 = {g2, g2, g2, g2, g2, g2, g2, g2};
        v8f c1 = c0;

        v8f d0 = __builtin_amdgcn_wmma_f32_16x16x4_f32(
            false, a0, false, bm, (short)0, c0, false, false);
        v8f d1 = __builtin_amdgcn_wmma_f32_16x16x4_f32(
            false, a1, false, bm, (short)0, c1, false, false);

        float colmin = BIG_F;
#pragma unroll
        for (int r = 0; r < 8; ++r) {
            fmin0[r] = fminf(fmin0[r], d0[r]);    // running fwd min, tile 0 rows
            fmin1[r] = fminf(fmin1[r], d1[r]);    // running fwd min, tile 1 rows
            colmin   = fminf(colmin, fminf(d0[r], d1[r]));  // bwd over 16 rows
        }
        colmin = fmaxf(colmin, 0.0f);             // clamp cancellation negatives
        colmin = fminf(colmin, __shfl_xor(colmin, 16, 32)); // all 32 rows
        if (half == 0)
            atomicMin(&lmin[n], __float_as_uint(colmin));   // ds_min_u32
    }

    // ---- forward: reduce each accumulator component across its 16 lanes ----
#pragma unroll
    for (int r = 0; r < 8; ++r) {
        float v = fmin0[r];
        v = fminf(v, __shfl_xor(v, 1, 32));
        v = fminf(v, __shfl_xor(v, 2, 32));
        v = fminf(v, __shfl_xor(v, 4, 32));
        v = fminf(v, __shfl_xor(v, 8, 32));
        fmin0[r] = v;

        float w = fmin1[r];
        w = fminf(w, __shfl_xor(w, 1, 32));
        w = fminf(w, __shfl_xor(w, 2, 32));
        w = fminf(w, __shfl_xor(w, 4, 32));
        w = fminf(w, __shfl_xor(w, 8, 32));
        fmin1[r] = w;
    }
    if (l16 == 0) {                               // lane 0: rows m0+{0..7,16..23}
        float s = 0.0f;                           // lane 16: rows m0+{8..15,24..31}
#pragma unroll
        for (int r = 0; r < 8; ++r) {
            s += sqrtf(fmaxf(fmin0[r], 0.0f) + CH_EPS);
            s += sqrtf(fmaxf(fmin1[r], 0.0f) + CH_EPS);
        }
        atomicAdd(out, s * inv_bm);
    }

    // ---- backward: flush block-level LDS mins to global once --------------
    __syncthreads();
    unsigned int* bg = bwd_min + (size_t)b * N;
    for (int i = tid; i < N; i += 256)
        atomicMin(&bg[i], lmin[i]);               // global_atomic_min_u32
}

// ---- kernel 2: sum sqrt(bwd_min + eps) into the scalar output --------------
__global__ void chamfer_bwd_reduce(const unsigned int* __restrict__ bwd_min,
                                   int total, float inv_bn,
                                   float* __restrict__ out) {
    int i = blockIdx.x * blockDim.x + threadIdx.x;
    float s = 0.0f;
    if (i < total)
        s = sqrtf(__uint_as_float(bwd_min[i]) + CH_EPS) * inv_bn;
#pragma unroll
    for (int off = 16; off > 0; off >>= 1)
        s += __shfl_down(s, off, 32);
    if ((threadIdx.x & 31) == 0 && s != 0.0f)
        atomicAdd(out, s);
}

extern "C" void kernel_launch(void* const* d_in, const int* in_sizes, int n_in,
                              void* d_out, int out_size, void* d_ws, size_t ws_size,
                              hipStream_t stream) {
    const float* P = (const float*)d_in[0];   // predict_pc (B,3,M)
    const float* G = (const float*)d_in[1];   // gt_pc      (B,3,N)
    float*       out = (float*)d_out;

    const int M = in_sizes[0] / (CH_B * CH_D);   // 8192
    const int N = in_sizes[1] / (CH_B * CH_D);   // 8192

    unsigned int* bwd = (unsigned int*)d_ws;     // B*N u32 (256 KB)
    const int total = CH_B * N;

    chamfer_init_kernel<<<(total + 255) / 256, 256, 0, stream>>>(bwd, total, out);

    const int strips = M >> 8;                   // 32 strips of 256 rows
    chamfer_wmma_kernel<<<CH_B * strips, 256, 0, stream>>>(
        P, G, M, N, bwd, out, 1.0f / ((float)CH_B * (float)M));

    chamfer_bwd_reduce<<<(total + 255) / 256, 256, 0, stream>>>(
        bwd, total, 1.0f / ((float)CH_B * (float)N), out);
}